// FittedRankOneRNN_18296560681337
// MI455X (gfx1250) — compile-verified
//
#include <hip/hip_runtime.h>
#include <math.h>

// Problem constants (match reference: B=256, T=1000, N=1024, dt/tau = 0.2)
#define RNN_B 256
#define RNN_T 1000
#define RNN_N 1024
#define RNN_ALPHA 0.2f

#if __has_builtin(__builtin_amdgcn_tanhf)
#define TANH_F32(x) __builtin_amdgcn_tanhf(x)
#elif __has_builtin(__builtin_amdgcn_tanh_f32)
#define TANH_F32(x) __builtin_amdgcn_tanh_f32(x)
#else
#define TANH_F32(x) tanhf(x)
#endif

typedef __attribute__((ext_vector_type(2))) float v2f;
typedef __attribute__((ext_vector_type(8))) float v8f;

// One wave32 per batch. Lane L owns elements j = (g*32+L)*4 + c, g<8, c<4
// (vectorized float4 loads; mapping is consistent across m/n/wi/w/x so the
// order-agnostic reductions are exact).
__global__ __launch_bounds__(32) void FittedRankOneRNN_kernel(
    const float* __restrict__ u,   // [B, T]
    const float* __restrict__ m,   // [N]
    const float* __restrict__ n,   // [N]
    const float* __restrict__ wi,  // [N]
    const float* __restrict__ w,   // [N]
    float* __restrict__ z)         // [B, T]
{
    const int b    = blockIdx.x;
    const int lane = threadIdx.x;  // 0..31 (wave32)

    // ---- load weights into registers (4 * 32 VGPRs) ----
    float mv[32], nv[32], wiv[32], wv[32], xs[32];
    const float4* m4  = reinterpret_cast<const float4*>(m);
    const float4* n4  = reinterpret_cast<const float4*>(n);
    const float4* wi4 = reinterpret_cast<const float4*>(wi);
    const float4* w4  = reinterpret_cast<const float4*>(w);
#pragma unroll
    for (int g = 0; g < 8; ++g) {
        const int vidx = g * 32 + lane;
        float4 a;
        a = m4[vidx];  mv[4*g+0]=a.x;  mv[4*g+1]=a.y;  mv[4*g+2]=a.z;  mv[4*g+3]=a.w;
        a = n4[vidx];  nv[4*g+0]=a.x;  nv[4*g+1]=a.y;  nv[4*g+2]=a.z;  nv[4*g+3]=a.w;
        a = wi4[vidx]; wiv[4*g+0]=a.x; wiv[4*g+1]=a.y; wiv[4*g+2]=a.z; wiv[4*g+3]=a.w;
        a = w4[vidx];  wv[4*g+0]=a.x;  wv[4*g+1]=a.y;  wv[4*g+2]=a.z;  wv[4*g+3]=a.w;
    }
#pragma unroll
    for (int i = 0; i < 32; ++i) xs[i] = 0.0f;  // x0 = 0

    // Constant selector B-matrix for the WMMA reduction:
    //   B[0,n]=B[2,n]=1 for n<8 else 0   (VGPR0: lanes hold K=0 / K=2)
    //   B[1,n]=B[3,n]=1 for n>=8 else 0  (VGPR1: lanes hold K=1 / K=3)
    v2f bsel;
    bsel.x = ((lane & 15) < 8) ? 1.0f : 0.0f;
    bsel.y = 1.0f - bsel.x;

    const float C0    = 1.0f - RNN_ALPHA;         // (1 - alpha)
    const float INV_N = 1.0f / (float)RNN_N;
    const float* ub   = u + (size_t)b * RNN_T;
    float* zb         = z + (size_t)b * RNN_T;

    float kappa = 0.0f;                            // tanh(0)@n/N = 0
    float uc_cur = (lane < RNN_T) ? ub[lane] : 0.0f;  // u chunk 0
    float zchunk = 0.0f;

    for (int tc = 0; tc < RNN_T; tc += 32) {
        // issue next u chunk load early (consumed 32 steps later)
        float uc_next = 0.0f;
        const int tn = tc + 32 + lane;
        if (tn < RNN_T) uc_next = ub[tn];

        const int nsteps = (RNN_T - tc) < 32 ? (RNN_T - tc) : 32;
#pragma unroll 1
        for (int s = 0; s < nsteps; ++s) {
            const float ut = __shfl(uc_cur, s, 32);
            const float ck = RNN_ALPHA * kappa;    // uniform across lanes
            const float cu = RNN_ALPHA * ut;

            // x <- (1-a)x + a*kappa*m + a*u*wi ; r = tanh(x);
            // accumulate sn = sum r*n, sw = sum r*w with split accumulators
            float sn0 = 0.f, sn1 = 0.f, sn2 = 0.f, sn3 = 0.f;
            float sw0 = 0.f, sw1 = 0.f, sw2 = 0.f, sw3 = 0.f;
#pragma unroll
            for (int i = 0; i < 32; i += 4) {
                float x0 = fmaf(C0, xs[i+0], fmaf(ck, mv[i+0], cu * wiv[i+0]));
                float x1 = fmaf(C0, xs[i+1], fmaf(ck, mv[i+1], cu * wiv[i+1]));
                float x2 = fmaf(C0, xs[i+2], fmaf(ck, mv[i+2], cu * wiv[i+2]));
                float x3 = fmaf(C0, xs[i+3], fmaf(ck, mv[i+3], cu * wiv[i+3]));
                xs[i+0] = x0; xs[i+1] = x1; xs[i+2] = x2; xs[i+3] = x3;
                const float r0 = TANH_F32(x0);
                const float r1 = TANH_F32(x1);
                const float r2 = TANH_F32(x2);
                const float r3 = TANH_F32(x3);
                sn0 = fmaf(r0, nv[i+0], sn0); sw0 = fmaf(r0, wv[i+0], sw0);
                sn1 = fmaf(r1, nv[i+1], sn1); sw1 = fmaf(r1, wv[i+1], sw1);
                sn2 = fmaf(r2, nv[i+2], sn2); sw2 = fmaf(r2, wv[i+2], sw2);
                sn3 = fmaf(r3, nv[i+3], sn3); sw3 = fmaf(r3, wv[i+3], sw3);
            }
            const float sn = (sn0 + sn1) + (sn2 + sn3);
            const float sw = (sw0 + sw1) + (sw2 + sw3);

            // ---- dual 32-lane reduction via one V_WMMA_F32_16X16X4_F32 ----
            // A VGPR0 = sn  => A[m,0]=sn[m],  A[m,2]=sn[m+16]
            // A VGPR1 = sw  => A[m,1]=sw[m],  A[m,3]=sw[m+16]
            // With the selector B: D[m,n<8] = sn[m]+sn[m+16],
            //                      D[m,n>=8]= sw[m]+sw[m+16]
            v2f a; a.x = sn; a.y = sw;
            v8f cz = {};
            v8f d = __builtin_amdgcn_wmma_f32_16x16x4_f32(
                false, a, false, bsel, (short)0, cz, false, false);
            // D VGPR v holds row M=v (lanes 0-15) and M=v+8 (lanes 16-31):
            // summing the 8 VGPRs gives per-lane: lanes 0-7  -> sum_{m<8}  p_m
            //   lanes 8-15 -> sum_{m<8} q_m ; lanes 16-23/24-31 -> m>=8 halves
            float acc = ((d[0] + d[1]) + (d[2] + d[3])) +
                        ((d[4] + d[5]) + (d[6] + d[7]));
            acc += __shfl_xor(acc, 16, 32);        // lanes 0-7: SN, 8-15: SW
            const float SN = __shfl(acc, 0, 32);
            const float SW = __shfl(acc, 8, 32);

            kappa = SN * INV_N;                    // kappa_{t+1}
            const float zt = SW * INV_N;           // z_t
            zchunk = (lane == s) ? zt : zchunk;    // stash for coalesced store
        }

        if (tc + lane < RNN_T) zb[tc + lane] = zchunk;
        uc_cur = uc_next;
    }
}

extern "C" void kernel_launch(void* const* d_in, const int* in_sizes, int n_in,
                              void* d_out, int out_size, void* d_ws, size_t ws_size,
                              hipStream_t stream) {
    const float* u  = (const float*)d_in[0];  // [B*T]
    const float* m  = (const float*)d_in[1];  // [N]
    const float* n  = (const float*)d_in[2];  // [N]
    const float* wi = (const float*)d_in[3];  // [N]
    const float* w  = (const float*)d_in[4];  // [N]
    float* z = (float*)d_out;                 // [B*T]
    (void)in_sizes; (void)n_in; (void)out_size; (void)d_ws; (void)ws_size;

    FittedRankOneRNN_kernel<<<RNN_B, 32, 0, stream>>>(u, m, n, wi, w, z);
}